// AttSampler_67405216744056
// MI455X (gfx1250) — compile-verified
//
#include <hip/hip_runtime.h>
#include <hip/hip_bf16.h>
#include <stdint.h>

// AttSampler for MI455X (gfx1250).
// Phase 1: water-fill + inverse-CDF coords (tiny, 32 blocks).
// Phase 2: separable bilinear grid_sample. One workgroup per (n, ho) output
// row; the two needed source rows (all 3 channels = 12KB) are staged into LDS
// via the CDNA5 async global->LDS path (ASYNCcnt), then lanes gather from LDS
// for the x interpolation. Memory-bound: ~360MB traffic total (~15us roofline
// at 23.3 TB/s). No contraction dimension anywhere -> WMMA inapplicable; the
// CDNA5 win is the async copy path + separable row staging.

#define NN   32
#define CC   3
#define HH   512
#define WW   512
#define AA   512
#define HO   512
#define WO   512

typedef int v4i __attribute__((ext_vector_type(4)));

// ---------- CDNA5 async global->LDS helpers (builtin if present, else asm) --
__device__ __forceinline__ void async_copy_b128(const float* g, float* l) {
#if __has_builtin(__builtin_amdgcn_global_load_async_to_lds_b128)
  v4i* gp = reinterpret_cast<v4i*>(const_cast<float*>(g));
  __builtin_amdgcn_global_load_async_to_lds_b128(
      (__attribute__((address_space(1))) v4i*)gp,
      (__attribute__((address_space(3))) v4i*)reinterpret_cast<v4i*>(l),
      0 /*imm offset*/, 0 /*cpol*/);
#else
  unsigned lds_off = (unsigned)(size_t)(__attribute__((address_space(3))) float*)l;
  asm volatile("global_load_async_to_lds_b128 %0, %1, off"
               :: "v"(lds_off), "v"(g) : "memory");
#endif
}

__device__ __forceinline__ void wait_async0() {
#if __has_builtin(__builtin_amdgcn_s_wait_asynccnt)
  __builtin_amdgcn_s_wait_asynccnt(0);
#else
  asm volatile("s_wait_asynccnt 0" ::: "memory");
#endif
}

// ---------------- Kernel 1: water-fill + coords ----------------------------
__global__ __launch_bounds__(512)
void k_waterfill_coords(const float* __restrict__ attx,
                        const float* __restrict__ atty,
                        float* __restrict__ gx,   // (N, WO) width coords
                        float* __restrict__ gy) { // (N, HO) height coords
  __shared__ float sax[AA], say[AA], scx[AA], scy[AA];
  const int n = blockIdx.x;
  const int i = threadIdx.x;

  float ax = attx[n * AA + i] * (float)HO;   // out_size = 512
  float ay = atty[n * AA + i] * (float)WO;   // out_sizey = 512

  for (int j = 0; j < 5; ++j) {
    // coupled max reduction
    scx[i] = ax; scy[i] = ay; __syncthreads();
    for (int s = 256; s > 0; s >>= 1) {
      if (i < s) { scx[i] = fmaxf(scx[i], scx[i + s]);
                   scy[i] = fmaxf(scy[i], scy[i + s]); }
      __syncthreads();
    }
    float thr = fminf(scx[0], scy[0]);
    if (j == 0) thr = fminf(thr, 4.0f);   // thr0 = SCALE*DENSE*H/A = 4
    __syncthreads();
    ax = fminf(ax, thr);
    ay = fminf(ay, thr);
    // sum reduction
    scx[i] = ax; scy[i] = ay; __syncthreads();
    for (int s = 256; s > 0; s >>= 1) {
      if (i < s) { scx[i] += scx[i + s]; scy[i] += scy[i + s]; }
      __syncthreads();
    }
    ax += ((float)HO - scx[0]) * (1.0f / (float)AA);
    ay += ((float)WO - scy[0]) * (1.0f / (float)AA);
    __syncthreads();
  }

  if (i == 0) { ax = 1.0f; ay = 1.0f; }   // first element forced to 1
  sax[i] = ax; say[i] = ay;
  scx[i] = ax; scy[i] = ay; __syncthreads();

  // inclusive scan (Hillis-Steele) -> cumsum in scx/scy
  for (int off = 1; off < AA; off <<= 1) {
    float vx = (i >= off) ? scx[i - off] : 0.0f;
    float vy = (i >= off) ? scy[i - off] : 0.0f;
    __syncthreads();
    scx[i] += vx; scy[i] += vy;
    __syncthreads();
  }

  // gy from (ax, csx): inverse-CDF via binary search (searchsorted left)
  {
    float step = scx[AA - 1] * (1.0f / (float)HO);
    float t = (float)(i + 1) * step;
    int lo = 0, hi = AA;
    while (lo < hi) { int mid = (lo + hi) >> 1;
                      if (scx[mid] < t) lo = mid + 1; else hi = mid; }
    int jj = lo < (AA - 1) ? lo : (AA - 1);
    float prev = (jj > 0) ? scx[jj - 1] : 0.0f;
    float frac = (t - prev) / fmaxf(sax[jj], 1e-12f);
    gy[n * HO + i] = ((float)jj + frac) * (2.0f / (float)AA) - 1.0f;
  }
  // gx from (ay, csy)
  {
    float step = scy[AA - 1] * (1.0f / (float)WO);
    float t = (float)(i + 1) * step;
    int lo = 0, hi = AA;
    while (lo < hi) { int mid = (lo + hi) >> 1;
                      if (scy[mid] < t) lo = mid + 1; else hi = mid; }
    int jj = lo < (AA - 1) ? lo : (AA - 1);
    float prev = (jj > 0) ? scy[jj - 1] : 0.0f;
    float frac = (t - prev) / fmaxf(say[jj], 1e-12f);
    gx[n * WO + i] = ((float)jj + frac) * (2.0f / (float)AA) - 1.0f;
  }
}

// ---------------- Kernel 2: separable bilinear grid_sample -----------------
// One block per (ho, n). y is constant across the row -> stage rows y0,y1 of
// all 3 channels into LDS with async global->LDS, then interpolate in x.
__global__ __launch_bounds__(256)
void k_grid_sample(const float* __restrict__ data,
                   const float* __restrict__ gx,
                   const float* __restrict__ gy,
                   float* __restrict__ sampled,  // (N,C,HO,WO)
                   float* __restrict__ grid) {   // (N,HO,WO,2)
  __shared__ __attribute__((aligned(16))) float rows[CC * 2 * WW]; // 12 KB
  const int ho = blockIdx.x;
  const int n  = blockIdx.y;
  const int t  = threadIdx.x;

  const float gyv = gy[n * HO + ho];
  const float y   = (gyv + 1.0f) * ((float)HH * 0.5f) - 0.5f;
  const float y0f = floorf(y);
  const int   y0  = (int)y0f, y1 = y0 + 1;
  float wy1 = y - y0f, wy0 = 1.0f - wy1;
  if (y0 < 0 || y0 >= HH) wy0 = 0.0f;         // zero-padding via weights
  if (y1 < 0 || y1 >= HH) wy1 = 0.0f;
  const int iy0 = min(max(y0, 0), HH - 1);
  const int iy1 = min(max(y1, 0), HH - 1);

  // Async-stage 6 rows (3 channels x {y0,y1}) -> LDS, 16B per lane per step.
  const float* base = data + (size_t)n * CC * HH * WW;
  #pragma unroll
  for (int k = 0; k < 3; ++k) {
    const int idx = t + k * 256;          // 0..767 vec4 chunks
    const int row = idx >> 7;             // 0..5
    const int off = (idx & 127) << 2;     // float offset within row
    const int c   = row >> 1;
    const int iy  = (row & 1) ? iy1 : iy0;
    const float* g = base + ((size_t)c * HH + iy) * WW + off;
    async_copy_b128(g, &rows[row * WW + off]);
  }
  wait_async0();
  __syncthreads();

  const size_t sBase = (size_t)n * CC * HO * WO + (size_t)ho * WO;
  float2* g2 = (float2*)grid + ((size_t)n * HO + ho) * WO;

  for (int wo = t; wo < WO; wo += 256) {
    const float gxv = gx[n * WO + wo];
    const float x   = (gxv + 1.0f) * ((float)WW * 0.5f) - 0.5f;
    const float x0f = floorf(x);
    const int   x0  = (int)x0f, x1 = x0 + 1;
    float wx1 = x - x0f, wx0 = 1.0f - wx1;
    if (x0 < 0 || x0 >= WW) wx0 = 0.0f;
    if (x1 < 0 || x1 >= WW) wx1 = 0.0f;
    const int ix0 = min(max(x0, 0), WW - 1);
    const int ix1 = min(max(x1, 0), WW - 1);

    #pragma unroll
    for (int c = 0; c < CC; ++c) {
      const float v00 = rows[(c * 2 + 0) * WW + ix0];
      const float v01 = rows[(c * 2 + 0) * WW + ix1];
      const float v10 = rows[(c * 2 + 1) * WW + ix0];
      const float v11 = rows[(c * 2 + 1) * WW + ix1];
      const float v = wy0 * (wx0 * v00 + wx1 * v01)
                    + wy1 * (wx0 * v10 + wx1 * v11);
      sampled[sBase + (size_t)c * HO * WO + wo] = v;
    }
    g2[wo] = make_float2(gxv, gyv);
  }
}

extern "C" void kernel_launch(void* const* d_in, const int* in_sizes, int n_in,
                              void* d_out, int out_size, void* d_ws, size_t ws_size,
                              hipStream_t stream) {
  const float* data = (const float*)d_in[0];   // (32,3,512,512)
  const float* attx = (const float*)d_in[1];   // (32,512)
  const float* atty = (const float*)d_in[2];   // (32,512)

  float* out     = (float*)d_out;
  float* sampled = out;                                    // 32*3*512*512
  float* grid    = out + (size_t)NN * CC * HO * WO;        // 32*512*512*2

  float* gx = (float*)d_ws;            // (N, WO)
  float* gy = gx + (size_t)NN * WO;    // (N, HO)   total 128 KB scratch

  k_waterfill_coords<<<NN, 512, 0, stream>>>(attx, atty, gx, gy);

  dim3 g2(HO, NN);
  k_grid_sample<<<g2, 256, 0, stream>>>(data, gx, gy, sampled, grid);
}